// newMM_77180562309386
// MI455X (gfx1250) — compile-verified
//
#include <hip/hip_runtime.h>
#include <hip/hip_bf16.h>

typedef float v2f __attribute__((ext_vector_type(2)));
typedef float v4f __attribute__((ext_vector_type(4)));
typedef float v8f __attribute__((ext_vector_type(8)));

#define WAVES_PER_BLOCK 8

// Kernel 1: per wave, handle 16 rows of the (B*8, 30) sign matrix:
//   - compute signs into LDS (A tile, K padded to 32)
//   - WMMA f32 16x16x4 against H (in LDS, 32x256) for 16 N-tiles
//   - running argmax over 256 columns, cross-lane reduce, write idx
__global__ __launch_bounds__(256) void sign_score_argmax_kernel(
    const float* __restrict__ x, const float* __restrict__ S,
    const float* __restrict__ T, const float* __restrict__ H,
    int* __restrict__ idxOut)
{
    __shared__ float Hs[32 * 256];                     // 32 KB, K rows 30,31 zero
    __shared__ float As[WAVES_PER_BLOCK][16 * 33];     // 16 rows x 32 K, pad 33 (bank-conflict-free)

    const int tid  = threadIdx.x;
    const int lane = tid & 31;
    const int wave = tid >> 5;

    // ---- load H (30x256) into LDS, zero-pad K to 32 ----
    for (int i = tid; i < 32 * 256; i += 256) {
        int d = i >> 8, n = i & 255;
        Hs[i] = (d < 30) ? H[d * 256 + n] : 0.0f;
    }

    const int gwave = blockIdx.x * WAVES_PER_BLOCK + wave;
    const int base  = gwave * 16;          // first global row of (B*8, 30)
    const int b0    = base >> 3;           // first batch index (wave covers b0, b0+1)

    // ---- compute signs: lane owns column d of A for all 16 rows ----
    const int d = lane;
    if (d < 30) {
        const int q = (d >= 15) ? 1 : 0;   // which half of the 30-wide group
        const int k = d - 15 * q;          // inner index into S/T (fixed per lane)
#pragma unroll
        for (int c = 0; c < 8; ++c) {
            const int c2 = 2 * c + q;                 // channel 0..15
            const float s0 = S[c2 * 30 + k];          // S[c2,0,k]
            const float s1 = S[c2 * 30 + 15 + k];     // S[c2,1,k]
            const float tt = T[c2 * 15 + k];
#pragma unroll
            for (int bb = 0; bb < 2; ++bb) {
                const int b = b0 + bb;
                const float x0 = x[b * 32 + 2 * c2];
                const float x1 = x[b * 32 + 2 * c2 + 1];
                float hv = x0 * s0 + x1 * s1;
                hv = hv - tt;
                hv = hv - 1e-4f;
                const float sg = (hv > 0.0f) ? 1.0f : ((hv < 0.0f) ? -1.0f : 0.0f);
                As[wave][(bb * 8 + c) * 33 + d] = sg;
            }
        }
    } else {
#pragma unroll
        for (int r = 0; r < 16; ++r) As[wave][r * 33 + d] = 0.0f;
    }

    __syncthreads();

    // ---- hoist A fragments: lane = (m = lane&15, K-pair at 4*kk + 2*(lane>>4)) ----
    const int m      = lane & 15;
    const int khalf2 = (lane >> 4) * 2;
    const int ncol   = lane & 15;
    const float* Arow = &As[wave][m * 33];

    v2f aFrag[8];
#pragma unroll
    for (int kk = 0; kk < 8; ++kk) {
        const int d0 = kk * 4 + khalf2;
        aFrag[kk][0] = Arow[d0];
        aFrag[kk][1] = Arow[d0 + 1];
    }

    float bestV[8];
    int   bestI[8];
#pragma unroll
    for (int v = 0; v < 8; ++v) { bestV[v] = -3.0e38f; bestI[v] = 0; }

    // ---- 16 N-tiles x 8 K-steps of V_WMMA_F32_16X16X4_F32 ----
    for (int nt = 0; nt < 16; ++nt) {
        v8f acc = {0.0f, 0.0f, 0.0f, 0.0f, 0.0f, 0.0f, 0.0f, 0.0f};
        const int n = nt * 16 + ncol;
#pragma unroll
        for (int kk = 0; kk < 8; ++kk) {
            const int d0 = kk * 4 + khalf2;
            v2f bFrag;
            bFrag[0] = Hs[d0 * 256 + n];
            bFrag[1] = Hs[(d0 + 1) * 256 + n];
            acc = __builtin_amdgcn_wmma_f32_16x16x4_f32(
                false, aFrag[kk], false, bFrag, (short)0, acc, false, false);
        }
        // D layout: this lane holds column n for rows v (lanes 0-15) / v+8 (lanes 16-31)
#pragma unroll
        for (int v = 0; v < 8; ++v) {
            const float s = acc[v];
            if (s > bestV[v]) { bestV[v] = s; bestI[v] = n; }
        }
    }

    // ---- argmax reduce across the 16 lanes of each half (first-max tie-break) ----
#pragma unroll
    for (int msk = 1; msk < 16; msk <<= 1) {
#pragma unroll
        for (int v = 0; v < 8; ++v) {
            const float ov = __shfl_xor(bestV[v], msk, 32);
            const int   oi = __shfl_xor(bestI[v], msk, 32);
            if (ov > bestV[v] || (ov == bestV[v] && oi < bestI[v])) {
                bestV[v] = ov; bestI[v] = oi;
            }
        }
    }

    if ((lane & 15) == 0) {
        const int roff = (lane >> 4) * 8;   // lanes 0..15 -> rows 0..7; 16..31 -> rows 8..15
#pragma unroll
        for (int v = 0; v < 8; ++v) idxOut[base + roff + v] = bestI[v];
    }
}

// Kernel 2: out[r,:] = LUT[r&7, idx[r], :]; pure bandwidth, NT stores.
__global__ __launch_bounds__(128) void gather_lut_kernel(
    const int* __restrict__ idx, const float* __restrict__ LUT,
    float* __restrict__ out, int nclass)
{
    const int r  = blockIdx.x;
    const int c  = r & 7;
    const int id = idx[r];                                   // uniform -> scalar load
    const v4f* __restrict__ src =
        (const v4f*)(LUT + ((size_t)c * 256 + (size_t)id) * (size_t)nclass);
    v4f* __restrict__ dst = (v4f*)(out + (size_t)r * (size_t)nclass);
    const int nv = nclass >> 2;                              // 250 vec4 per row
    for (int t = threadIdx.x; t < nv; t += 128) {
        v4f v = src[t];
        __builtin_nontemporal_store(v, &dst[t]);
    }
}

extern "C" void kernel_launch(void* const* d_in, const int* in_sizes, int n_in,
                              void* d_out, int out_size, void* d_ws, size_t ws_size,
                              hipStream_t stream) {
    const float* x   = (const float*)d_in[0];   // (B, 32)
    const float* S   = (const float*)d_in[1];   // (16, 2, 15)
    const float* H   = (const float*)d_in[2];   // (30, 256)
    const float* T   = (const float*)d_in[3];   // (16, 15)
    const float* LUT = (const float*)d_in[4];   // (8, 256, nclass)
    float* out = (float*)d_out;

    const int Bsz    = in_sizes[0] / 32;        // 16384
    const int R      = Bsz * 8;                 // 131072 rows of (B*8, 30)
    const int nclass = out_size / R;            // 1000

    int* idxbuf = (int*)d_ws;                   // R ints = 512 KB scratch

    const int rowsPerBlock = 16 * WAVES_PER_BLOCK;          // 128
    const int nblocks1 = (R + rowsPerBlock - 1) / rowsPerBlock;
    sign_score_argmax_kernel<<<nblocks1, 256, 0, stream>>>(x, S, T, H, idxbuf);

    gather_lut_kernel<<<R, 128, 0, stream>>>(idxbuf, LUT, out, nclass);
}